// Rotation_71150428226260
// MI455X (gfx1250) — compile-verified
//
#include <hip/hip_runtime.h>

// ---------------------------------------------------------------------------
// Local qudit rotation:  out = (I_27 ⊗ M ⊗ I_81) · x,  x complex (6561, 512)
// M = [[c, -i s, 0], [-i s, c, 0], [0,0,1]],  c = cos(θ/2), s = sin(θ/2)
//
// Real 6×6 form T, component order k = [x0re, x1re, x2re, x0im, x1im, x2im]:
//   out0re =  c·x0re + s·x1im        out0im =  c·x0im - s·x1re
//   out1re =  c·x1re + s·x0im        out1im =  c·x1im - s·x0re
//   out2re =  x2re                   out2im =  x2im
//
// Bandwidth bound: 54 MB @ 23.3 TB/s ≈ 2.3 µs floor; compute is free, so the
// 6×6 is done with V_WMMA_F32_16X16X4_F32 (K = 4+2, zero padded):
//   A (16×4) = constant T chunk, synthesized per lane from (c,s) — no memory
//   B (4×16) = 16 state columns; every 16-lane load = one 64B contiguous row
//   D rows 0..5 = the 6 output components for 16 columns -> 64B stores
//
// Round-2 changes:
//   * trig hoisted to a 1-wave setup kernel (angle is uniform) -> the hot
//     kernel loses the ~300-instr sin/cos argument-reduction preamble and
//     just scalar-loads (c, s) from d_ws.
//   * 32 columns per wave (2 adjacent groups, shared A, 4 WMMAs) -> half the
//     per-wave overhead, denser vmem clauses (group 1 = +64B immediates).
// ---------------------------------------------------------------------------

typedef __attribute__((ext_vector_type(2))) float v2f;
typedef __attribute__((ext_vector_type(8))) float v8f;

#define RIGHT   81                         // 3^(L-1-INDEX)
#define BCOLS   512                        // batch columns
#define NSTATE  6561                       // 3^8
#define NB_ELT  ((size_t)NSTATE * BCOLS)   // elements per real/imag plane
#define PAIRS   (27 * 81)                  // (l, r) pairs = 2187
#define CGRPS   (BCOLS / 32)               // 32-column tiles per pair = 16
#define TILES   (PAIRS * CGRPS)            // 34992 tiles, one per wave
#define WAVES_PER_BLOCK 8                  // 256 threads, wave32

// --- setup: compute cos/sin of the uniform angle once ----------------------
__global__ void trig_setup_kernel(const float* __restrict__ ang,
                                  float* __restrict__ cs) {
    if (threadIdx.x == 0) {
        const float th = 0.5f * ang[0];
        cs[0] = cosf(th);
        cs[1] = sinf(th);
    }
}

// --- main streaming kernel -------------------------------------------------
__global__ __launch_bounds__(256)
void rot_wmma_kernel(const float* __restrict__ xr,
                     const float* __restrict__ xi,
                     const float* __restrict__ cs,
                     float* __restrict__ out)
{
    const int lane = threadIdx.x & 31;
    const int m    = lane & 15;          // A row index / B column index
    const int hi   = lane >> 4;          // half-wave selector (K pairing)
    const int wid  = blockIdx.x * WAVES_PER_BLOCK + (threadIdx.x >> 5);

    // tile -> (l, r) pair + 32-column group; consecutive waves stream
    // consecutive 128B column spans of the same rows.
    const int cg   = wid & (CGRPS - 1);
    const int pair = wid >> 4;
    const int l    = pair / RIGHT;
    const int r    = pair - l * RIGHT;
    const int r0   = l * (3 * RIGHT) + r;        // global row of component j=0
    const int col  = cg * 32 + m;                // group-0 column (group 1: +16)

    // uniform scalar loads (precomputed by trig_setup_kernel)
    const float c = cs[0];
    const float s = cs[1];

    // ---- A operand: constant T chunks (ISA 16x4 f32 A layout:
    //      VGPR0 = A[m][K=0] lanes 0-15 / A[m][K=2] lanes 16-31;
    //      VGPR1 = A[m][K=1] / A[m][K=3]).
    // chunk0 columns: k0=x0re k1=x1re k2=x2re k3=x0im
    const float a0k0 = (m == 0) ? c : (m == 4) ? -s : 0.0f;
    const float a0k1 = (m == 1) ? c : (m == 3) ? -s : 0.0f;
    const float a0k2 = (m == 2) ? 1.0f : 0.0f;
    const float a0k3 = (m == 1) ? s : (m == 3) ? c : 0.0f;
    // chunk1 columns: k0=x1im k1=x2im (k2,k3 zero padding)
    const float a1k0 = (m == 0) ? s : (m == 4) ? c : 0.0f;
    const float a1k1 = (m == 5) ? 1.0f : 0.0f;

    v2f A0, A1;
    A0.x = hi ? a0k2 : a0k0;
    A0.y = hi ? a0k3 : a0k1;
    A1.x = hi ? 0.0f : a1k0;
    A1.y = hi ? 0.0f : a1k1;

    // ---- B operands for both 16-column groups (group 1 = col + 16).
    //      Layout mirrors A: VGPR0 = B[0][n] lanes 0-15 / B[2][n] lanes 16-31;
    //      VGPR1 = B[1][n] / B[3][n]. Each 16-lane segment is 64B contiguous.
    const size_t row02 = (size_t)(r0 + (hi ? 2 * RIGHT : 0)) * BCOLS + col;
    v2f B0a, B0b;
    B0a.x = xr[row02];                          // B[0]=x0re / B[2]=x2re
    B0b.x = xr[row02 + 16];
    {
        const float* p   = hi ? xi : xr;        // B[1]=x1re / B[3]=x0im
        const size_t rw  = (size_t)(hi ? r0 : (r0 + RIGHT)) * BCOLS + col;
        B0a.y = p[rw];
        B0b.y = p[rw + 16];
    }
    v2f B1a = {0.0f, 0.0f}, B1b = {0.0f, 0.0f}; // chunk1: lanes 0-15 only
    if (!hi) {
        const size_t r1 = (size_t)(r0 + RIGHT)     * BCOLS + col;
        const size_t r2 = (size_t)(r0 + 2 * RIGHT) * BCOLS + col;
        B1a.x = xi[r1];                          // x1im
        B1b.x = xi[r1 + 16];
        B1a.y = xi[r2];                          // x2im
        B1b.y = xi[r2 + 16];
    }

    // ---- D = A0*B0 + A1*B1 per group (chained accumulate). EXEC is all-1s:
    // the divergent ifs above are closed before the convergent WMMA ops.
    v8f acca = {}, accb = {};
    acca = __builtin_amdgcn_wmma_f32_16x16x4_f32(false, A0, false, B0a,
                                                 (short)0, acca, false, false);
    accb = __builtin_amdgcn_wmma_f32_16x16x4_f32(false, A0, false, B0b,
                                                 (short)0, accb, false, false);
    acca = __builtin_amdgcn_wmma_f32_16x16x4_f32(false, A1, false, B1a,
                                                 (short)0, acca, false, false);
    accb = __builtin_amdgcn_wmma_f32_16x16x4_f32(false, A1, false, B1b,
                                                 (short)0, accb, false, false);

    // ---- Stores: D VGPR v, lanes 0-15 hold component v of the 16 columns.
    // Twelve 64B contiguous segments; lanes 16-31 (zero rows 8..13) masked.
    if (!hi) {
        const size_t base = (size_t)r0 * BCOLS + col;
        const size_t rs   = (size_t)RIGHT * BCOLS;
        out[base]                        = acca[0];   // out0re, cols 0-15
        out[base + 16]                   = accb[0];   // out0re, cols 16-31
        out[base + rs]                   = acca[1];   // out1re
        out[base + rs + 16]              = accb[1];
        out[base + 2 * rs]               = acca[2];   // out2re
        out[base + 2 * rs + 16]          = accb[2];
        out[NB_ELT + base]               = acca[3];   // out0im
        out[NB_ELT + base + 16]          = accb[3];
        out[NB_ELT + base + rs]          = acca[4];   // out1im
        out[NB_ELT + base + rs + 16]     = accb[4];
        out[NB_ELT + base + 2 * rs]      = acca[5];   // out2im
        out[NB_ELT + base + 2 * rs + 16] = accb[5];
    }
}

extern "C" void kernel_launch(void* const* d_in, const int* in_sizes, int n_in,
                              void* d_out, int out_size, void* d_ws, size_t ws_size,
                              hipStream_t stream) {
    const float* xr  = (const float*)d_in[0];   // x_real (N, B) f32
    const float* xi  = (const float*)d_in[1];   // x_imag (N, B) f32
    const float* ang = (const float*)d_in[2];   // angle  (1,)   f32
    float* out = (float*)d_out;                 // (2, N, B) f32
    float* cs  = (float*)d_ws;                  // {cos(θ/2), sin(θ/2)}

    trig_setup_kernel<<<1, 32, 0, stream>>>(ang, cs);

    const int blocks = TILES / WAVES_PER_BLOCK; // 4374, exact
    rot_wmma_kernel<<<blocks, 256, 0, stream>>>(xr, xi, cs, out);
}